// ControlNetSpatialEmbedder_8409545965710
// MI455X (gfx1250) — compile-verified
//
#include <hip/hip_runtime.h>

// Paint 1024 boxes (last-writer-wins) into (5, 2048, 2048) f32.
// Store-BW bound: 84 MB out / 23.3 TB/s ~ 3.6 us floor; input 16 KB -> LDS.
// CDNA5 paths: global_load_async_to_lds_b128 (+ s_wait_asynccnt), wave32
// ballot early-exit, LDS atomics for per-tile box mask, non-temporal b128
// stores (output is write-once, never re-read -> keep it out of L2).

#define HH 2048
#define WW 2048
#define NB 1024
#define TILE 32
#define NTHREADS 256

typedef __attribute__((ext_vector_type(4))) float v4f;
typedef __attribute__((ext_vector_type(4))) int   v4i;

__global__ __launch_bounds__(NTHREADS)
void paint_boxes_kernel(const float* __restrict__ boxes, float* __restrict__ out) {
    __shared__ v4f      boxF[NB];      // raw boxes (ymin,xmin,ymax,xmax), 16 KB
    __shared__ v4i      bnds[NB];      // integer pixel bounds (y1,x1,y2,x2), 16 KB
    __shared__ unsigned mask[NB / 32]; // boxes intersecting this tile

    const int tid = threadIdx.x;

    // ---- Stage the 16 KB box table into LDS via async DMA (ASYNCcnt) ----
    // 256 threads x 4 B128 transfers = 1024 * 16 B.
    {
        unsigned           lds0 = (unsigned)(size_t)(&boxF[0]); // LDS byte offset
        unsigned long long g0   = (unsigned long long)(size_t)boxes;
#pragma unroll
        for (int i = 0; i < 4; ++i) {
            unsigned           loff = lds0 + (unsigned)(tid * 16 + i * 4096);
            unsigned long long ga   = g0 + (unsigned long long)(tid * 16 + i * 4096);
            asm volatile("global_load_async_to_lds_b128 %0, %1, off"
                         :: "v"(loff), "v"(ga)
                         : "memory");
        }
        asm volatile("s_wait_asynccnt 0" ::: "memory");
    }
    if (tid < NB / 32) mask[tid] = 0u;
    __syncthreads();

    const int tx0 = blockIdx.x * TILE;
    const int ty0 = blockIdx.y * TILE;

    // ---- Per-box integer bounds + tile-intersection mask ----
    for (int b = tid; b < NB; b += NTHREADS) {
        v4f f  = boxF[b];
        int y1 = max(0, (int)(f.x * (float)HH)); // trunc like .astype(int32)
        int x1 = max(0, (int)(f.y * (float)WW));
        int y2 = min(HH, (int)(f.z * (float)HH));
        int x2 = min(WW, (int)(f.w * (float)WW));
        bnds[b] = (v4i){y1, x1, y2, x2};
        bool hit = (y1 < y2) & (x1 < x2) &
                   (y1 < ty0 + TILE) & (y2 > ty0) &
                   (x1 < tx0 + TILE) & (x2 > tx0);
        if (hit) atomicOr(&mask[b >> 5], 1u << (b & 31));
    }
    __syncthreads();

    // ---- Per-pixel search: highest-index covering box wins ----
    // Thread owns 4 consecutive x pixels of one row: x0 = tx0 + (tid%8)*4,
    // y = ty0 + tid/8. 8 lanes * 16 B = 128 B contiguous per wave row.
    const int x0 = tx0 + ((tid & 7) << 2);
    const int y  = ty0 + (tid >> 3);

    int  best[4] = { -1, -1, -1, -1 };
    bool all_done = false;
    for (int w = (NB / 32) - 1; w >= 0 && !all_done; --w) {
        unsigned m = mask[w]; // wave-uniform word
        while (m) {
            int b = 31 - __builtin_clz(m); // highest set bit first (descending index)
            m &= ~(1u << b);
            int box = (w << 5) | b;
            v4i bb  = bnds[box]; // wave-uniform address -> LDS broadcast
            if ((y >= bb.x) & (y < bb.z)) {
#pragma unroll
                for (int k = 0; k < 4; ++k) {
                    int xx = x0 + k;
                    if (best[k] < 0 && xx >= bb.y && xx < bb.w) best[k] = box;
                }
            }
            bool undecided = (best[0] < 0) | (best[1] < 0) | (best[2] < 0) | (best[3] < 0);
            if (__builtin_amdgcn_ballot_w32(undecided) == 0u) { all_done = true; break; }
        }
    }

    // ---- Vectorized non-temporal channel stores (zeros if uncovered) ----
    float c0[4], c1[4], c2[4], c3[4], c4[4];
#pragma unroll
    for (int k = 0; k < 4; ++k) {
        int b = best[k];
        if (b >= 0) {
            v4f f = boxF[b];
            c0[k] = 1.0f; c1[k] = f.x; c2[k] = f.y; c3[k] = f.z; c4[k] = f.w;
        } else {
            c0[k] = 0.f; c1[k] = 0.f; c2[k] = 0.f; c3[k] = 0.f; c4[k] = 0.f;
        }
    }
    const size_t HWs = (size_t)HH * (size_t)WW;
    const size_t idx = (size_t)y * WW + (size_t)x0; // 16-B aligned (x0 % 4 == 0)
    __builtin_nontemporal_store((v4f){c0[0], c0[1], c0[2], c0[3]}, (v4f*)(out + idx));
    __builtin_nontemporal_store((v4f){c1[0], c1[1], c1[2], c1[3]}, (v4f*)(out + HWs + idx));
    __builtin_nontemporal_store((v4f){c2[0], c2[1], c2[2], c2[3]}, (v4f*)(out + 2 * HWs + idx));
    __builtin_nontemporal_store((v4f){c3[0], c3[1], c3[2], c3[3]}, (v4f*)(out + 3 * HWs + idx));
    __builtin_nontemporal_store((v4f){c4[0], c4[1], c4[2], c4[3]}, (v4f*)(out + 4 * HWs + idx));
}

extern "C" void kernel_launch(void* const* d_in, const int* in_sizes, int n_in,
                              void* d_out, int out_size, void* d_ws, size_t ws_size,
                              hipStream_t stream) {
    (void)in_sizes; (void)n_in; (void)out_size; (void)d_ws; (void)ws_size;
    const float* boxes = (const float*)d_in[0];
    float*       out   = (float*)d_out;
    dim3 grid(WW / TILE, HH / TILE); // 64 x 64 tiles
    paint_boxes_kernel<<<grid, NTHREADS, 0, stream>>>(boxes, out);
}